// LearnableSineOscillator_67740224193122
// MI455X (gfx1250) — compile-verified
//
#include <hip/hip_runtime.h>

typedef __attribute__((ext_vector_type(2))) float v2f;
typedef __attribute__((ext_vector_type(8))) float v8f;

#define WAVE_SIZE        32
#define WAVES_PER_BLOCK  8
#define BLOCK_SIZE       (WAVE_SIZE * WAVES_PER_BLOCK)   // 256
#define CHUNK            256                             // outputs per WMMA
#define CHUNKS_PER_WAVE  8
#define ELEMS_PER_WAVE   (CHUNK * CHUNKS_PER_WAVE)       // 2048
#define ELEMS_PER_BLOCK  (ELEMS_PER_WAVE * WAVES_PER_BLOCK) // 16384

// y[k] = Re(p0 * zc^k),  zc = z * tanh(|z|)/|z|
// Tile k = k0 + c*256 + 16*i + j  (i,j in 0..15):
//   zc^k = (p0 * zc^(k0+c*256+16i)) * zc^j = A_i * B_j   (complex)
//   Re(A_i*B_j) = ReA_i*ReB_j + ImA_i*(-ImB_j)           -> rank-2 outer product
// -> one V_WMMA_F32_16X16X4_F32 per 256 outputs (K rows 0,1 used; 2,3 zero).
__global__ __launch_bounds__(BLOCK_SIZE)
void osc_wmma_kernel(const float* __restrict__ zrp, const float* __restrict__ zip,
                     const float* __restrict__ prp, const float* __restrict__ pip,
                     float* __restrict__ out, unsigned n)
{
    const float zr = zrp[0], zi = zip[0];
    const float pr = prp[0], pi = pip[0];

    const float mag = sqrtf(zr * zr + zi * zi);
    const float r   = tanhf(mag);          // |zc| (constrain)
    const float lnr = __logf(r);
    const float th  = atan2f(zi, zr);      // arg(zc) == arg(z)

    const unsigned lane = threadIdx.x & (WAVE_SIZE - 1);
    const unsigned wave = blockIdx.x * WAVES_PER_BLOCK + (threadIdx.x >> 5);
    const unsigned k0   = wave * ELEMS_PER_WAVE;
    if (k0 >= n) return;                   // wave-uniform: EXEC stays all-1s

    const bool  low = lane < 16;
    const float jf  = (float)(lane & 15);

    // B column j: (Re zc^j, -Im zc^j) in K rows 0,1; K rows 2,3 (lanes 16-31) zero.
    float bm = __expf(lnr * jf);
    float bs, bc; __sincosf(th * jf, &bs, &bc);
    v2f B;
    B.x = low ?  bm * bc  : 0.0f;
    B.y = low ? -(bm * bs) : 0.0f;

    // A row i: p0 * zc^(k0 + 16*i) in K cols 0,1; K cols 2,3 zero.
    // k0+16i < 2^23 -> exact in f32. Distant waves underflow to exact 0 (matches cumprod).
    float kA = (float)k0 + 16.0f * jf;
    float am = __expf(lnr * kA);
    float as, ac; __sincosf(th * kA, &as, &ac);
    float ar0 = am * ac, ai0 = am * as;
    float Ar = pr * ar0 - pi * ai0;
    float Ai = pr * ai0 + pi * ar0;
    v2f A;
    A.x = low ? Ar : 0.0f;
    A.y = low ? Ai : 0.0f;

    // Per-chunk advance factor zc^256 (uniform).
    float wm = __expf(lnr * 256.0f);
    float ws, wc; __sincosf(th * 256.0f, &ws, &wc);
    ws *= wm; wc *= wm;

    // D layout: VGPR v, lane l -> M = v + (l<16?0:8), N = l&15, k = base + M*16 + N
    const unsigned laneOff = ((lane >> 4) * 8) * 16 + (lane & 15);

    if (k0 + ELEMS_PER_WAVE <= n) {
        // Fast path (every wave when n % ELEMS_PER_WAVE == 0): unconditional NT stores,
        // EXEC stays all-1s throughout -> no v_cmpx/exec juggling around the WMMAs.
        for (unsigned c = 0; c < CHUNKS_PER_WAVE; ++c) {
            v8f acc = {};
            v8f d = __builtin_amdgcn_wmma_f32_16x16x4_f32(
                false, A, false, B, (short)0, acc, false, false);

            float* p = out + (k0 + c * CHUNK + laneOff);
#pragma unroll
            for (int v = 0; v < 8; ++v)
                __builtin_nontemporal_store(d[v], p + v * 16);

            // A *= zc^256 (complex, uniform multiplier)
            float nr = A.x * wc - A.y * ws;
            float ni = A.x * ws + A.y * wc;
            A.x = nr; A.y = ni;
        }
    } else {
        // Tail wave (generic n): per-lane guarded stores; WMMA still sees full EXEC
        // since the guard diverges only around the stores.
        for (unsigned c = 0; c < CHUNKS_PER_WAVE; ++c) {
            v8f acc = {};
            v8f d = __builtin_amdgcn_wmma_f32_16x16x4_f32(
                false, A, false, B, (short)0, acc, false, false);

            const unsigned base = k0 + c * CHUNK + laneOff;
#pragma unroll
            for (int v = 0; v < 8; ++v) {
                unsigned idx = base + (unsigned)v * 16u;
                if (idx < n)
                    __builtin_nontemporal_store(d[v], out + idx);
            }
            float nr = A.x * wc - A.y * ws;
            float ni = A.x * ws + A.y * wc;
            A.x = nr; A.y = ni;
        }
    }
}

extern "C" void kernel_launch(void* const* d_in, const int* in_sizes, int n_in,
                              void* d_out, int out_size, void* d_ws, size_t ws_size,
                              hipStream_t stream) {
    // setup_inputs order: x (unused, shape only), z_real, z_imag, phase_real, phase_imag, t (unused)
    const float* zrp = (const float*)d_in[1];
    const float* zip = (const float*)d_in[2];
    const float* prp = (const float*)d_in[3];
    const float* pip = (const float*)d_in[4];
    float* out = (float*)d_out;
    unsigned n = (unsigned)out_size;

    unsigned grid = (n + ELEMS_PER_BLOCK - 1) / ELEMS_PER_BLOCK;  // 512 for N=2^23
    osc_wmma_kernel<<<grid, BLOCK_SIZE, 0, stream>>>(zrp, zip, prp, pip, out, n);
}